// GraphModel_59133109732151
// MI455X (gfx1250) — compile-verified
//
#include <hip/hip_runtime.h>

#define N_NODES 50000
#define N_EDGES 800000
#define D_FEAT  96
#define HIDDEN  256
#define OUT_F   40

typedef __attribute__((ext_vector_type(2))) float v2f;
typedef __attribute__((ext_vector_type(8))) float v8f;

// ---------------------------------------------------------------------------
// Zero a float region (grid-stride).
// ---------------------------------------------------------------------------
__global__ void zero_kernel(float* __restrict__ p, int n) {
    int i = blockIdx.x * blockDim.x + threadIdx.x;
    int stride = gridDim.x * blockDim.x;
    for (; i < n; i += stride) p[i] = 0.0f;
}

// ---------------------------------------------------------------------------
// One 96-thread block per edge: gather features[src], atomic-add into
// summed[dst], lane 0 bumps the degree counter.
// ---------------------------------------------------------------------------
__global__ __launch_bounds__(96) void scatter_kernel(
    const float* __restrict__ feat, const int* __restrict__ src,
    const int* __restrict__ dst, float* __restrict__ summed,
    float* __restrict__ deg) {
    int e = blockIdx.x;
    int s = src[e];
    int d = dst[e];
    float v = feat[s * D_FEAT + threadIdx.x];
    atomicAdd(&summed[d * D_FEAT + threadIdx.x], v);
    if (threadIdx.x == 0) atomicAdd(&deg[d], 1.0f);
}

// ---------------------------------------------------------------------------
// combined[i, 0:96]   = features[i]
// combined[i, 96:192] = summed[i] / max(deg[i], 1)
// ---------------------------------------------------------------------------
__global__ void combine_kernel(const float* __restrict__ feat,
                               const float* __restrict__ summed,
                               const float* __restrict__ deg,
                               float* __restrict__ combined) {
    const int total = N_NODES * (2 * D_FEAT);
    int i = blockIdx.x * blockDim.x + threadIdx.x;
    int stride = gridDim.x * blockDim.x;
    for (; i < total; i += stride) {
        int node = i / (2 * D_FEAT);
        int c    = i % (2 * D_FEAT);
        float v;
        if (c < D_FEAT) {
            v = feat[node * D_FEAT + c];
        } else {
            v = summed[node * D_FEAT + (c - D_FEAT)] / fmaxf(deg[node], 1.0f);
        }
        combined[i] = v;
    }
}

// ---------------------------------------------------------------------------
// WMMA f32 GEMM: C[M,N] = act(A[M,K] @ B[K,N] + bias), all row-major.
// Block = 128 threads = 4 waves. Each wave computes a 64x16 output panel
// (4 accumulator tiles); the block covers 256x16 using V_WMMA_F32_16X16X4_F32.
//
// The block's 16-column B panel (K x 16) is staged once into LDS (<=16 KB),
// shared by all 4 waves; inner-loop B reads become conflict-free ds_loads
// (the two B values of a fragment are 64 B apart -> ds_load_2addr pattern).
//
// Fragment layouts (ISA 7.12.2, wave32):
//   A 16x4 : lanes 0-15 / 16-31 -> M rows; half 0 holds K={0,1} in {v0,v1},
//            half 1 holds K={2,3}.   (one aligned 8-byte load per tile)
//   B 4x16 : lanes 0-15 -> N cols; half 0 rows K={0,1}, half 1 rows K={2,3}.
//   C/D    : VGPR r: lanes 0-15 -> M=r, lanes 16-31 -> M=8+r, N=lane%16.
// ---------------------------------------------------------------------------
template <bool RELU>
__global__ __launch_bounds__(128) void wmma_gemm_kernel(
    const float* __restrict__ A, const float* __restrict__ B,
    const float* __restrict__ bias, float* __restrict__ C,
    int M, int K, int N) {
    __shared__ float Bs[HIDDEN * 16];   // K <= 256 -> up to 16 KB

    const int tid  = threadIdx.x;
    const int lane = tid & 31;
    const int wave = tid >> 5;
    const int half = lane >> 4;   // 0: K pair {0,1}, 1: K pair {2,3}
    const int l16  = lane & 15;

    const int nBase = blockIdx.x * 16;
    const int mBase = (blockIdx.y * 4 + wave) * 64;

    // ---- stage B panel [K x 16] into LDS (coalesced, column-clamped) ----
    const int panel = K * 16;
    for (int idx = tid; idx < panel; idx += 128) {
        int k = idx >> 4;
        int c = idx & 15;
        int bc = nBase + c;
        if (bc >= N) bc = N - 1;          // clamp (junk cols never stored)
        Bs[idx] = B[k * N + bc];
    }
    __syncthreads();

    const int col = nBase + l16;

    // Clamp row indices for safe loads (stores are guarded exactly).
    const int m0 = min(mBase +  0 + l16, M - 1);
    const int m1 = min(mBase + 16 + l16, M - 1);
    const int m2 = min(mBase + 32 + l16, M - 1);
    const int m3 = min(mBase + 48 + l16, M - 1);

    const float* A0 = A + m0 * K + 2 * half;   // 8-byte aligned (K even)
    const float* A1 = A + m1 * K + 2 * half;
    const float* A2 = A + m2 * K + 2 * half;
    const float* A3 = A + m3 * K + 2 * half;
    const float* Bl = Bs + (2 * half) * 16 + l16;

    v8f acc0 = {}, acc1 = {}, acc2 = {}, acc3 = {};

#pragma unroll 4
    for (int k = 0; k < K; k += 4) {
        v2f bf;
        bf.x = Bl[k * 16];         // row k + 2*half      (LDS)
        bf.y = Bl[k * 16 + 16];    // row k + 2*half + 1  (LDS, +64 B)

        v2f a0 = *(const v2f*)(A0 + k);
        v2f a1 = *(const v2f*)(A1 + k);
        v2f a2 = *(const v2f*)(A2 + k);
        v2f a3 = *(const v2f*)(A3 + k);

        acc0 = __builtin_amdgcn_wmma_f32_16x16x4_f32(false, a0, false, bf,
                                                     (short)0, acc0, false, false);
        acc1 = __builtin_amdgcn_wmma_f32_16x16x4_f32(false, a1, false, bf,
                                                     (short)0, acc1, false, false);
        acc2 = __builtin_amdgcn_wmma_f32_16x16x4_f32(false, a2, false, bf,
                                                     (short)0, acc2, false, false);
        acc3 = __builtin_amdgcn_wmma_f32_16x16x4_f32(false, a3, false, bf,
                                                     (short)0, acc3, false, false);
    }

    if (col < N) {
        const float bv = bias[col];
        const v8f accs[4] = {acc0, acc1, acc2, acc3};
#pragma unroll
        for (int t = 0; t < 4; ++t) {
#pragma unroll
            for (int r = 0; r < 8; ++r) {
                int row = mBase + t * 16 + half * 8 + r;
                if (row < M) {
                    float v = accs[t][r] + bv;
                    if (RELU) v = fmaxf(v, 0.0f);
                    C[row * N + col] = v;
                }
            }
        }
    }
}

// ---------------------------------------------------------------------------
// Launcher.  Inputs: features, src, dst, W1, b1, W2, b2, W3, b3.
// Workspace layout (bytes):
//   [0, 38.4M)       combined  [50000 x 192] f32
//   [38.4M, 89.6M)   h1        [50000 x 256] f32   (aliased early by summed+deg)
//   [89.6M, 140.8M)  h2        [50000 x 256] f32
// ---------------------------------------------------------------------------
extern "C" void kernel_launch(void* const* d_in, const int* in_sizes, int n_in,
                              void* d_out, int out_size, void* d_ws, size_t ws_size,
                              hipStream_t stream) {
    (void)in_sizes; (void)n_in; (void)out_size; (void)ws_size;

    const float* feat = (const float*)d_in[0];
    const int*   src  = (const int*)d_in[1];
    const int*   dst  = (const int*)d_in[2];
    const float* W1   = (const float*)d_in[3];
    const float* b1   = (const float*)d_in[4];
    const float* W2   = (const float*)d_in[5];
    const float* b2   = (const float*)d_in[6];
    const float* W3   = (const float*)d_in[7];
    const float* b3   = (const float*)d_in[8];
    float* out = (float*)d_out;

    char* ws = (char*)d_ws;
    float* combined = (float*)ws;                                       // 50000*192
    float* h1 = (float*)(ws + (size_t)N_NODES * 192 * sizeof(float));   // 50000*256
    float* h2 = (float*)(ws + (size_t)N_NODES * (192 + 256) * sizeof(float));

    // summed/deg alias the h1 region: they are dead before h1 is written.
    float* summed = h1;                              // 50000*96
    float* deg    = h1 + (size_t)N_NODES * D_FEAT;   // 50000 (contiguous)

    // 1) zero segment-sum accumulators + degrees (contiguous region)
    zero_kernel<<<1024, 256, 0, stream>>>(summed, N_NODES * (D_FEAT + 1));

    // 2) edge gather + scatter-add
    scatter_kernel<<<N_EDGES, 96, 0, stream>>>(feat, src, dst, summed, deg);

    // 3) build [features | neighbor_mean]
    combine_kernel<<<2048, 256, 0, stream>>>(feat, summed, deg, combined);

    // 4) GEMM1: [50000x192]@[192x256] + b1, ReLU
    dim3 blk(128);
    dim3 g1(HIDDEN / 16, (N_NODES + 255) / 256);
    wmma_gemm_kernel<true><<<g1, blk, 0, stream>>>(combined, W1, b1, h1,
                                                   N_NODES, 2 * D_FEAT, HIDDEN);

    // 5) GEMM2: [50000x256]@[256x256] + b2, ReLU
    dim3 g2(HIDDEN / 16, (N_NODES + 255) / 256);
    wmma_gemm_kernel<true><<<g2, blk, 0, stream>>>(h1, W2, b2, h2,
                                                   N_NODES, HIDDEN, HIDDEN);

    // 6) GEMM3: [50000x256]@[256x40] + b3 (no ReLU) -> d_out
    dim3 g3((OUT_F + 15) / 16, (N_NODES + 255) / 256);
    wmma_gemm_kernel<false><<<g3, blk, 0, stream>>>(h2, W3, b3, out,
                                                    N_NODES, HIDDEN, OUT_F);
}